// Decoder_31069793419876
// MI455X (gfx1250) — compile-verified
//
#include <hip/hip_runtime.h>
#include <math.h>

// Problem constants (from reference)
#define B_   64
#define S_   128
#define E_   512
#define H_   1024
#define V_   50257
#define XK_  (E_ + H_)     // 1536, GRU input width
#define K2_  (2 * H_)      // 2048, output-GEMM K

#define KC_  256           // K-chunk staged in LDS for logits GEMM
#define RS_  (KC_ + 4)     // padded LDS row stride (floats) -> conflict-free b64 reads
#define NTILES_ ((V_ + 15) / 16)        // 3142
#define GX_     ((NTILES_ + 7) / 8)     // 393 blocks in N

typedef __attribute__((ext_vector_type(2))) float v2f;
typedef __attribute__((ext_vector_type(8))) float v8f;
typedef __attribute__((ext_vector_type(4))) int   v4i;

// D = A(16x4) * B(4x16) + C, fp32. (neg_a, A, neg_b, B, c_mod, C, reuse_a, reuse_b)
__device__ __forceinline__ v8f wmma_f32(v2f a, v2f b, v8f c) {
    return __builtin_amdgcn_wmma_f32_16x16x4_f32(false, a, false, b, (short)0, c, false, false);
}

__device__ __forceinline__ float sigmoidf_(float v) { return 1.0f / (1.0f + expf(-v)); }

// ---- async global->LDS path (gfx1250), guarded --------------------------------
#if __has_builtin(__builtin_amdgcn_global_load_async_to_lds_b128) && \
    __has_builtin(__builtin_amdgcn_s_wait_asynccnt)
#define USE_ASYNC_LDS 1
typedef __attribute__((address_space(1))) v4i* as1_v4i;
typedef __attribute__((address_space(3))) v4i* as3_v4i;
__device__ __forceinline__ void cp16_async(const float* g, float* l) {
    __builtin_amdgcn_global_load_async_to_lds_b128((as1_v4i)g, (as3_v4i)l, 0, 0);
}
#endif

// ---------------------------------------------------------------------------
// K1: x = [emb_table[input_ids] | prev_context]   -> [B, 1536]
// ---------------------------------------------------------------------------
__global__ void build_x_kernel(const int* __restrict__ ids,
                               const float* __restrict__ emb,
                               const float* __restrict__ prev_ctx,
                               float* __restrict__ x) {
    int t = blockIdx.x * blockDim.x + threadIdx.x;
    int idx = t * 4;
    if (idx >= B_ * XK_) return;
    int b = idx / XK_;
    int c = idx - b * XK_;
    const float* src = (c < E_) ? (emb + (size_t)ids[b] * E_ + c)
                                : (prev_ctx + (size_t)b * H_ + (c - E_));
    *(float4*)(x + idx) = *(const float4*)src;
}

// ---------------------------------------------------------------------------
// K2: fused GRU cell. One wave per 16(M over B) x 16(N over H) tile.
// ---------------------------------------------------------------------------
__global__ void __launch_bounds__(32)
gru_kernel(const float* __restrict__ x, const float* __restrict__ hprev,
           const float* __restrict__ Wih, const float* __restrict__ Whh,
           const float* __restrict__ bih, const float* __restrict__ bhh,
           float* __restrict__ hnew) {
    const int lane = threadIdx.x;
    const int half = lane >> 4;
    const int l    = lane & 15;
    const int n0 = blockIdx.x * 16;   // over H
    const int m0 = blockIdx.y * 16;   // over B

    v8f ar{}, az{}, agn{}, ahn{};

    {   // gx = x @ W_ih.T (K = 1536)
        const float* xr = x   + (size_t)(m0 + l) * XK_;
        const float* w0 = Wih + (size_t)(0 * H_ + n0 + l) * XK_;
        const float* w1 = Wih + (size_t)(1 * H_ + n0 + l) * XK_;
        const float* w2 = Wih + (size_t)(2 * H_ + n0 + l) * XK_;
        for (int k = 0; k < XK_; k += 4) {
            const int kk = k + 2 * half;
            v2f a  = *(const v2f*)(xr + kk);
            v2f b0 = *(const v2f*)(w0 + kk);
            v2f b1 = *(const v2f*)(w1 + kk);
            v2f b2 = *(const v2f*)(w2 + kk);
            ar  = wmma_f32(a, b0, ar);
            az  = wmma_f32(a, b1, az);
            agn = wmma_f32(a, b2, agn);
        }
    }
    {   // gh = h_prev @ W_hh.T (K = 1024); gh_n kept separate (r gates it)
        const float* hr = hprev + (size_t)(m0 + l) * H_;
        const float* u0 = Whh + (size_t)(0 * H_ + n0 + l) * H_;
        const float* u1 = Whh + (size_t)(1 * H_ + n0 + l) * H_;
        const float* u2 = Whh + (size_t)(2 * H_ + n0 + l) * H_;
        for (int k = 0; k < H_; k += 4) {
            const int kk = k + 2 * half;
            v2f a  = *(const v2f*)(hr + kk);
            v2f b0 = *(const v2f*)(u0 + kk);
            v2f b1 = *(const v2f*)(u1 + kk);
            v2f b2 = *(const v2f*)(u2 + kk);
            ar  = wmma_f32(a, b0, ar);
            az  = wmma_f32(a, b1, az);
            ahn = wmma_f32(a, b2, ahn);
        }
    }

    const int nc = n0 + l;
    const float br  = bih[nc]          + bhh[nc];
    const float bz  = bih[H_ + nc]     + bhh[H_ + nc];
    const float bgn = bih[2 * H_ + nc];
    const float bhn = bhh[2 * H_ + nc];

    #pragma unroll
    for (int i = 0; i < 8; i++) {
        const int m = m0 + i + 8 * half;          // C/D layout: M = i + 8*half
        const float r = sigmoidf_(ar[i] + br);
        const float z = sigmoidf_(az[i] + bz);
        const float n = tanhf((agn[i] + bgn) + r * (ahn[i] + bhn));
        const float hp = hprev[(size_t)m * H_ + nc];
        hnew[(size_t)m * H_ + nc] = (1.0f - z) * n + z * hp;
    }
}

// ---------------------------------------------------------------------------
// K3: q = h_new @ W_attn   ([64,1024] x [1024,1024]),  B[k,n] = W_attn[k,n]
// ---------------------------------------------------------------------------
__global__ void __launch_bounds__(32)
qproj_kernel(const float* __restrict__ hnew, const float* __restrict__ Wattn,
             float* __restrict__ q) {
    const int lane = threadIdx.x;
    const int half = lane >> 4;
    const int l    = lane & 15;
    const int n0 = blockIdx.x * 16;
    const int m0 = blockIdx.y * 16;

    v8f acc{};
    const float* hr = hnew + (size_t)(m0 + l) * H_;
    for (int k = 0; k < H_; k += 4) {
        const int kk = k + 2 * half;
        v2f a = *(const v2f*)(hr + kk);
        v2f b; b.x = Wattn[(size_t)kk * H_ + n0 + l];
               b.y = Wattn[(size_t)(kk + 1) * H_ + n0 + l];
        acc = wmma_f32(a, b, acc);
    }
    #pragma unroll
    for (int i = 0; i < 8; i++) {
        const int m = m0 + i + 8 * half;
        q[(size_t)m * H_ + n0 + l] = acc[i];
    }
}

// ---------------------------------------------------------------------------
// K4: per-batch attention: scores = enc . q, mask, softmax, context
// ---------------------------------------------------------------------------
__global__ void __launch_bounds__(128)
attn_kernel(const float* __restrict__ q, const float* __restrict__ enc,
            const unsigned char* __restrict__ mask,
            float* __restrict__ attn_out, float* __restrict__ ctx_out) {
    __shared__ float qs[H_];
    __shared__ float sc[S_];
    __shared__ float tmp[128];

    const int b   = blockIdx.x;
    const int tid = threadIdx.x;
    const int w    = tid >> 5;
    const int lane = tid & 31;

    for (int h = tid; h < H_; h += 128) qs[h] = q[(size_t)b * H_ + h];
    __syncthreads();

    for (int j = 0; j < S_ / 4; j++) {
        const int s = w + 4 * j;
        const float* er = enc + ((size_t)b * S_ + s) * H_;
        float p = 0.0f;
        for (int c = lane * 4; c < H_; c += 128) {
            float4 e  = *(const float4*)(er + c);
            float4 qq = *(const float4*)(qs + c);
            p += e.x * qq.x + e.y * qq.y + e.z * qq.z + e.w * qq.w;
        }
        #pragma unroll
        for (int off = 16; off > 0; off >>= 1) p += __shfl_xor(p, off, 32);
        if (lane == 0) sc[s] = mask[(size_t)b * S_ + s] ? p : -1e9f;
    }
    __syncthreads();

    const float v = sc[tid];
    tmp[tid] = v; __syncthreads();
    for (int st = 64; st > 0; st >>= 1) {
        if (tid < st) tmp[tid] = fmaxf(tmp[tid], tmp[tid + st]);
        __syncthreads();
    }
    const float mx = tmp[0]; __syncthreads();
    const float e = expf(v - mx);
    tmp[tid] = e; __syncthreads();
    for (int st = 64; st > 0; st >>= 1) {
        if (tid < st) tmp[tid] += tmp[tid + st];
        __syncthreads();
    }
    const float a = e / tmp[0];
    sc[tid] = a;
    attn_out[(size_t)b * S_ + tid] = a;
    __syncthreads();

    float acc[8] = {0};
    for (int s = 0; s < S_; s++) {
        const float as = sc[s];
        const float* er = enc + ((size_t)b * S_ + s) * H_;
        #pragma unroll
        for (int r = 0; r < 8; r++) acc[r] += as * er[tid + 128 * r];
    }
    #pragma unroll
    for (int r = 0; r < 8; r++) ctx_out[(size_t)b * H_ + tid + 128 * r] = acc[r];
}

// ---------------------------------------------------------------------------
// K5: logits = [h_new | context] @ W_out.T + b_out   (the 412 MB HBM streamer)
// Block = 4 waves: one 16-row M-tile x 8 N-tiles. A tile staged through LDS in
// double-buffered 256-wide K-chunks (async global->LDS when available), each
// wave runs two WMMA accumulation chains so global VMEM is a pure W_out stream.
// ---------------------------------------------------------------------------
__global__ void __launch_bounds__(128)
logits_kernel(const float* __restrict__ hnew, const float* __restrict__ ctx,
              const float* __restrict__ Wout, const float* __restrict__ bout,
              float* __restrict__ logits) {
    __shared__ float abuf[2][16 * RS_];

    const int tid  = threadIdx.x;
    const int w    = tid >> 5;
    const int lane = tid & 31;
    const int half = lane >> 4;
    const int l    = lane & 15;
    const int m0   = blockIdx.y * 16;
    const int nt0  = blockIdx.x * 8 + 2 * w;

    const int n_a = nt0 * 16 + l;
    const int n_b = (nt0 + 1) * 16 + l;
    const int ra  = (n_a < V_) ? n_a : (V_ - 1);   // clamp: EXEC stays full
    const int rb  = (n_b < V_) ? n_b : (V_ - 1);
    const float* wrA = Wout + (size_t)ra * K2_;
    const float* wrB = Wout + (size_t)rb * K2_;

    v8f acc0{}, acc1{};

    // stage chunk c (k0 = c*KC_) of A = [h_new | ctx] rows m0..m0+15 into abuf[buf]
    auto stage = [&](int c, int buf) {
        const float* base = (c < 4) ? (hnew + (size_t)m0 * H_ + c * KC_)
                                    : (ctx  + (size_t)m0 * H_ + (c - 4) * KC_);
        #pragma unroll
        for (int j = 0; j < 8; j++) {
            const int idx = j * 512 + tid * 4;     // 4096 floats per chunk
            const int row = idx >> 8;
            const int col = idx & 255;
            const float* g = base + (size_t)row * H_ + col;
            float*       d = &abuf[buf][row * RS_ + col];
#ifdef USE_ASYNC_LDS
            cp16_async(g, d);
#else
            *(float4*)d = *(const float4*)g;
#endif
        }
    };

#ifdef USE_ASYNC_LDS
    stage(0, 0);
#endif
    for (int c = 0; c < 8; c++) {
#ifdef USE_ASYNC_LDS
        if (c + 1 < 8) {                               // uniform condition
            stage(c + 1, (c + 1) & 1);                 // prefetch next chunk
            __builtin_amdgcn_s_wait_asynccnt(8);       // chunk c's 8 ops done
        } else {
            __builtin_amdgcn_s_wait_asynccnt(0);       // drain for last chunk
        }
        __syncthreads();                               // all waves' copies visible
#else
        stage(c, c & 1);
        __syncthreads();
#endif
        const float* ab = abuf[c & 1];
        const float* wA = wrA + c * KC_;
        const float* wB = wrB + c * KC_;
        for (int k = 0; k < KC_; k += 4) {
            const int kk = k + 2 * half;
            v2f a  = *(const v2f*)&ab[l * RS_ + kk];   // conflict-free ds_load_b64
            v2f b0 = *(const v2f*)(wA + kk);
            v2f b1 = *(const v2f*)(wB + kk);
            acc0 = wmma_f32(a, b0, acc0);
            acc1 = wmma_f32(a, b1, acc1);
        }
        __syncthreads();   // buf[c&1] free for chunk c+2's staging
    }

    if (n_a < V_) {
        const float bo = bout[n_a];
        #pragma unroll
        for (int i = 0; i < 8; i++)
            logits[(size_t)(m0 + i + 8 * half) * V_ + n_a] = acc0[i] + bo;
    }
    if (n_b < V_) {
        const float bo = bout[n_b];
        #pragma unroll
        for (int i = 0; i < 8; i++)
            logits[(size_t)(m0 + i + 8 * half) * V_ + n_b] = acc1[i] + bo;
    }
}

// ---------------------------------------------------------------------------
// launcher
// ---------------------------------------------------------------------------
extern "C" void kernel_launch(void* const* d_in, const int* in_sizes, int n_in,
                              void* d_out, int out_size, void* d_ws, size_t ws_size,
                              hipStream_t stream) {
    const int*   ids      = (const int*)d_in[0];
    const float* hidden   = (const float*)d_in[1];   // [1,B,H] == [B,H]
    const float* prev_ctx = (const float*)d_in[2];
    const float* enc      = (const float*)d_in[3];
    const unsigned char* mask = (const unsigned char*)d_in[4];
    const float* emb      = (const float*)d_in[5];
    const float* Wattn    = (const float*)d_in[6];
    const float* Wih      = (const float*)d_in[7];
    const float* Whh      = (const float*)d_in[8];
    const float* bih      = (const float*)d_in[9];
    const float* bhh      = (const float*)d_in[10];
    const float* Wout     = (const float*)d_in[11];
    const float* bout     = (const float*)d_in[12];

    float* out    = (float*)d_out;                   // tuple concat, flat:
    float* logits = out;                             // [B,V]
    float* hnew   = out + (size_t)B_ * V_;           // [1,B,H]
    float* ctx    = hnew + (size_t)B_ * H_;          // [B,H]
    float* attn   = ctx  + (size_t)B_ * H_;          // [B,S]

    float* ws = (float*)d_ws;
    float* x  = ws;                                  // [B, 1536]
    float* q  = ws + (size_t)B_ * XK_;               // [B, H]

    build_x_kernel<<<(B_ * XK_ / 4 + 255) / 256, 256, 0, stream>>>(ids, emb, prev_ctx, x);
    gru_kernel<<<dim3(H_ / 16, B_ / 16), 32, 0, stream>>>(x, hidden, Wih, Whh, bih, bhh, hnew);
    qproj_kernel<<<dim3(H_ / 16, B_ / 16), 32, 0, stream>>>(hnew, Wattn, q);
    attn_kernel<<<B_, 128, 0, stream>>>(q, enc, mask, attn, ctx);
    logits_kernel<<<dim3(GX_, B_ / 16), 128, 0, stream>>>(hnew, ctx, Wout, bout, logits);
}